// PUPHAWUnsupervisedLoss_25709674234593
// MI455X (gfx1250) — compile-verified
//
#include <hip/hip_runtime.h>
#include <stdint.h>

// ---------------------------------------------------------------------------
// Memory-bound graph-loss pipeline for MI455X (gfx1250, wave32).
//  - fp32 global atomics for all edge scatters (random dst -> L2 atomic BW)
//  - sorted-row SpMV with per-thread run compaction (~2 atomics / 64 nnz)
//  - wave reductions via V_WMMA_F32_16X16X4_F32 (ones-B trick) + shfl_xor(16)
// ---------------------------------------------------------------------------

typedef __attribute__((ext_vector_type(2))) float v2f;
typedef __attribute__((ext_vector_type(8))) float v8f;

#define BLOCK 256
#define SPMV_CHUNK 64

// Wave32 sum via WMMA: A.x = v (lanes 0..15 -> A[m][0], lanes 16..31 -> A[m][2]),
// A.y = 0, B = all ones. D[m][n] = v_m + v_{m+16}. Sum D's 8 VGPRs per lane,
// then fold lanes 0..15 with 16..31. EXEC must be all ones (callers guarantee).
__device__ __forceinline__ float wave_reduce_sum_wmma(float v) {
  v2f a; a.x = v;    a.y = 0.0f;
  v2f b; b.x = 1.0f; b.y = 1.0f;
  v8f c = {0.f, 0.f, 0.f, 0.f, 0.f, 0.f, 0.f, 0.f};
  v8f d = __builtin_amdgcn_wmma_f32_16x16x4_f32(
      /*neg_a=*/false, a, /*neg_b=*/false, b,
      /*c_mod=*/(short)0, c, /*reuse_a=*/false, /*reuse_b=*/false);
  float s = ((d[0] + d[1]) + (d[2] + d[3])) + ((d[4] + d[5]) + (d[6] + d[7]));
  s += __shfl_xor(s, 16);
  return s;  // full 32-lane sum in every lane
}

__global__ void zero_f32(float* p, long long n) {
  long long i = blockIdx.x * (long long)blockDim.x + threadIdx.x;
  if (i < n) p[i] = 0.0f;
}

// Pass 1 over edges: deg += 1, qsum += edge_len*|dpred|, net += (p_src - p_dst)
__global__ void edge_pass1(const long long* __restrict__ src,
                           const long long* __restrict__ dst,
                           const float* __restrict__ pred,
                           const float* __restrict__ feats,
                           float* __restrict__ deg, float* __restrict__ qsum,
                           float* __restrict__ net, long long E) {
  long long e = blockIdx.x * (long long)blockDim.x + threadIdx.x;
  if (e >= E) return;
  long long s = src[e];
  long long d = dst[e];
  float ps = pred[s], pd = pred[d];
  float2 cs = *(const float2*)(feats + 8 * s);   // feats[:, :2], row stride 8
  float2 cd = *(const float2*)(feats + 8 * d);
  float dx = cs.x - cd.x, dy = cs.y - cd.y;
  float len  = sqrtf(dx * dx + dy * dy + 1e-12f);
  float grad = fabsf(ps - pd);
  atomicAdd(&deg[d], 1.0f);
  atomicAdd(&qsum[d], len * grad);
  atomicAdd(&net[d], ps - pd);
}

// w0_raw = 1 + qsum/max(deg,1); clamp deg in place; block-partial sum of w0_raw
__global__ void node_w0(const float* __restrict__ qsum, float* __restrict__ deg,
                        float* __restrict__ wcell, float* __restrict__ cur,
                        float* __restrict__ sum_w0, int N) {
  int i = blockIdx.x * blockDim.x + threadIdx.x;
  float w = 0.0f;
  if (i < N) {
    float dg = fmaxf(deg[i], 1.0f);
    deg[i] = dg;                      // store clamped degree for hop passes
    w = 1.0f + qsum[i] / dg;
    wcell[i] = w;
    cur[i]   = w;
  }
  float s = wave_reduce_sum_wmma(w);  // all lanes active; OOB lanes contribute 0
  if ((threadIdx.x & 31) == 0) atomicAdd(sum_w0, s);
}

// w0 /= mean(w0)  <=>  *= N / sum
__global__ void node_scale(float* __restrict__ wcell, float* __restrict__ cur,
                           const float* __restrict__ sum_w0, int N) {
  int i = blockIdx.x * blockDim.x + threadIdx.x;
  if (i >= N) return;
  float scale = (float)N / sum_w0[0];
  wcell[i] *= scale;
  cur[i]   *= scale;
}

// Hop scatter: nxt[dst] += cur[src]
__global__ void edge_hop(const long long* __restrict__ src,
                         const long long* __restrict__ dst,
                         const float* __restrict__ cur,
                         float* __restrict__ nxt, long long E) {
  long long e = blockIdx.x * (long long)blockDim.x + threadIdx.x;
  if (e >= E) return;
  atomicAdd(&nxt[dst[e]], cur[src[e]]);
}

// cur = nxt/deg; wcell += decay^h * cur; reset nxt for next hop
__global__ void node_hop(float* __restrict__ cur, float* __restrict__ nxt,
                         const float* __restrict__ deg, float* __restrict__ wcell,
                         float decay_pow, int N) {
  int i = blockIdx.x * blockDim.x + threadIdx.x;
  if (i >= N) return;
  float c = nxt[i] / deg[i];
  nxt[i] = 0.0f;
  cur[i] = c;
  wcell[i] += decay_pow * c;
}

// SpMV with sorted row_idx: fold runs locally, atomicAdd only at run breaks.
__global__ void spmv_sorted(const float* __restrict__ vals,
                            const long long* __restrict__ col,
                            const long long* __restrict__ row,
                            const float* __restrict__ pred,
                            float* __restrict__ Ax, long long E) {
  long long t = blockIdx.x * (long long)blockDim.x + threadIdx.x;
  long long start = t * (long long)SPMV_CHUNK;
  if (start >= E) return;
  long long end = start + SPMV_CHUNK;
  if (end > E) end = E;
  long long curRow = row[start];
  float acc = 0.0f;
  for (long long j = start; j < end; ++j) {
    long long r = row[j];
    if (r != curRow) {
      atomicAdd(&Ax[curRow], acc);
      curRow = r;
      acc = 0.0f;
    }
    acc += vals[j] * pred[col[j]];
  }
  atomicAdd(&Ax[curRow], acc);
}

// Per-node loss terms -> 10 wave-WMMA reductions -> global accumulators
__global__ void final_pass(const float* __restrict__ pred,
                           const float* __restrict__ b,
                           const float* __restrict__ Ax,
                           const float* __restrict__ wcell,
                           const float* __restrict__ net,
                           const unsigned char* __restrict__ inlet,
                           const unsigned char* __restrict__ outlet,
                           const unsigned char* __restrict__ wall,
                           const unsigned char* __restrict__ icm,
                           const float* __restrict__ icv,
                           float* __restrict__ acc, int N) {
  int i = blockIdx.x * blockDim.x + threadIdx.x;
  float q[10];
#pragma unroll
  for (int k = 0; k < 10; ++k) q[k] = 0.0f;
  if (i < N) {
    float p = pred[i];
    float r = Ax[i] - b[i];
    q[0] = wcell[i] * r * r;                                 // pde numerator
    if (inlet[i])  { float d = p - 0.1f;   q[1] = d * d; q[2] = 1.0f; }
    if (outlet[i]) {                        q[3] = p * p; q[4] = 1.0f; }
    if (wall[i])   {                        q[5] = p * p; q[6] = 1.0f; }
    if (icm[i])    { float d = p - icv[i]; q[7] = d * d; q[8] = 1.0f; }
    float nv = net[i];
    q[9] = nv * nv;                                          // cons numerator
  }
#pragma unroll
  for (int k = 0; k < 10; ++k) {
    float s = wave_reduce_sum_wmma(q[k]);   // EXEC all-ones, no early return
    if ((threadIdx.x & 31) == 0) atomicAdd(&acc[k], s);
  }
}

__global__ void combine(const float* __restrict__ acc, float* __restrict__ out,
                        int N) {
  if (blockIdx.x == 0 && threadIdx.x == 0) {
    float invN = 1.0f / (float)N;
    float total = acc[0] * invN                          // loss_pde
                + acc[1] / fmaxf(acc[2], 1.0f)           // inlet
                + acc[3] / fmaxf(acc[4], 1.0f)           // outlet
                + acc[5] / fmaxf(acc[6], 1.0f)           // wall
                + acc[7] / fmaxf(acc[8], 1.0f)           // ic
                + acc[9] * invN;                         // loss_cons
    out[0] = total;
  }
}

extern "C" void kernel_launch(void* const* d_in, const int* in_sizes, int n_in,
                              void* d_out, int out_size, void* d_ws, size_t ws_size,
                              hipStream_t stream) {
  const float*         pred   = (const float*)d_in[0];
  const float*         feats  = (const float*)d_in[1];
  const float*         bvec   = (const float*)d_in[2];
  const float*         vals   = (const float*)d_in[3];
  const float*         icv    = (const float*)d_in[4];
  // d_in[5] = row_ptr (unused by reference)
  const long long*     col    = (const long long*)d_in[6];
  const long long*     row    = (const long long*)d_in[7];
  const long long*     eidx   = (const long long*)d_in[8];   // [2, E]
  const unsigned char* icm    = (const unsigned char*)d_in[9];
  const unsigned char* inlet  = (const unsigned char*)d_in[10];
  const unsigned char* outlet = (const unsigned char*)d_in[11];
  const unsigned char* wall   = (const unsigned char*)d_in[12];
  // d_in[13] = epoch (unused by reference)

  const int       N = in_sizes[0];
  const long long E = (long long)in_sizes[6];

  // Workspace layout (floats): [0]=sum_w0, [1..10]=loss accumulators,
  // then 7 node arrays of N floats each (needs (16 + 7N)*4 bytes ~ 2.8 MB).
  float* ws     = (float*)d_ws;
  float* sum_w0 = ws;
  float* acc    = ws + 1;
  float* deg    = ws + 16;
  float* qsum   = deg + N;
  float* net    = qsum + N;
  float* wcell  = net + N;
  float* cur    = wcell + N;
  float* nxt    = cur + N;
  float* Ax     = nxt + N;

  const long long src_off = 0, dst_off = E;
  const int nb = (N + BLOCK - 1) / BLOCK;
  const int eb = (int)((E + BLOCK - 1) / BLOCK);

  long long zn = 16 + 7LL * N;
  zero_f32<<<(int)((zn + BLOCK - 1) / BLOCK), BLOCK, 0, stream>>>(ws, zn);

  edge_pass1<<<eb, BLOCK, 0, stream>>>(eidx + src_off, eidx + dst_off, pred,
                                       feats, deg, qsum, net, E);

  node_w0<<<nb, BLOCK, 0, stream>>>(qsum, deg, wcell, cur, sum_w0, N);
  node_scale<<<nb, BLOCK, 0, stream>>>(wcell, cur, sum_w0, N);

  // NUM_HOPS = 2, DECAY = 0.5 -> decay powers 0.5, 0.25
  for (int h = 0; h < 2; ++h) {
    edge_hop<<<eb, BLOCK, 0, stream>>>(eidx + src_off, eidx + dst_off, cur, nxt, E);
    node_hop<<<nb, BLOCK, 0, stream>>>(cur, nxt, deg, wcell,
                                       (h == 0) ? 0.5f : 0.25f, N);
  }

  const long long spmv_threads = (E + SPMV_CHUNK - 1) / SPMV_CHUNK;
  const int sb = (int)((spmv_threads + BLOCK - 1) / BLOCK);
  spmv_sorted<<<sb, BLOCK, 0, stream>>>(vals, col, row, pred, Ax, E);

  final_pass<<<nb, BLOCK, 0, stream>>>(pred, bvec, Ax, wcell, net, inlet, outlet,
                                       wall, icm, icv, acc, N);

  combine<<<1, 32, 0, stream>>>(acc, (float*)d_out, N);
}